// MambaPytorch_19576460935842
// MI455X (gfx1250) — compile-verified
//
#include <hip/hip_runtime.h>
#include <math.h>

// Model dims (fixed by the reference)
#define VOCAB 32000
#define D_    1024
#define NL_   4
#define DI_   2048
#define S_    16
#define R_    64
#define DC_   4
#define L_    1024
#define EPSF  1e-5f
#define DBCW  (R_ + 2 * S_)   // 96

typedef __attribute__((ext_vector_type(2))) float v2f;
typedef __attribute__((ext_vector_type(8))) float v8f;
typedef __attribute__((ext_vector_type(4))) unsigned int v4u;
typedef __attribute__((ext_vector_type(8))) int v8i;
typedef __attribute__((ext_vector_type(4))) int v4i;

#if defined(__has_builtin)
#if __has_builtin(__builtin_amdgcn_tensor_load_to_lds)
#define HAVE_TDM 1
#endif
#endif
#ifndef HAVE_TDM
#define HAVE_TDM 0
#endif

// ---------------------------------------------------------------------------
// Embedding gather: x[l, :] = embed[tokens[l], :]   (D=1024 -> 256 float4)
// ---------------------------------------------------------------------------
__global__ __launch_bounds__(256) void embed_gather_k(
    const int* __restrict__ tok, const float* __restrict__ emb,
    float* __restrict__ x) {
  const int l = blockIdx.x;
  const int t = tok[l];
  const float4* src = (const float4*)(emb + (size_t)t * D_);
  float4* dst = (float4*)(x + (size_t)l * D_);
  dst[threadIdx.x] = src[threadIdx.x];
}

// ---------------------------------------------------------------------------
// RMSNorm: one row per block, 256 threads, wave32 shfl + LDS reduction
// ---------------------------------------------------------------------------
__global__ __launch_bounds__(256) void rmsnorm_k(
    const float* __restrict__ x, const float* __restrict__ w,
    float* __restrict__ o) {
  const int l = blockIdx.x;
  const float* row = x + (size_t)l * D_;
  float p = 0.f;
  for (int i = threadIdx.x; i < D_; i += 256) {
    float v = row[i];
    p += v * v;
  }
  for (int off = 16; off > 0; off >>= 1) p += __shfl_xor(p, off, 32);
  __shared__ float sp[8];
  if ((threadIdx.x & 31) == 0) sp[threadIdx.x >> 5] = p;
  __syncthreads();
  float total = 0.f;
#pragma unroll
  for (int i = 0; i < 8; ++i) total += sp[i];
  const float scale = rsqrtf(total / (float)D_ + EPSF);
  for (int i = threadIdx.x; i < D_; i += 256)
    o[(size_t)l * D_ + i] = row[i] * scale * w[i];
}

// ---------------------------------------------------------------------------
// Causal depthwise conv1d (DC=4) + bias + SiLU on u = xz[:, :DI]
// ---------------------------------------------------------------------------
__global__ __launch_bounds__(256) void conv_silu_k(
    const float* __restrict__ xz, const float* __restrict__ cw,
    const float* __restrict__ cb, float* __restrict__ u) {
  const int d = blockIdx.x * 256 + threadIdx.x;
  const int l = blockIdx.y;
  float acc = cb[d];
#pragma unroll
  for (int j = 0; j < DC_; ++j) {
    const int ll = l - (DC_ - 1) + j;
    if (ll >= 0) acc += cw[d * DC_ + j] * xz[(size_t)ll * (2 * DI_) + d];
  }
  u[(size_t)l * DI_ + d] = acc / (1.f + __expf(-acc));  // silu
}

// ---------------------------------------------------------------------------
// Selective scan: thread d owns 16 states in registers; sequential over L.
// ---------------------------------------------------------------------------
__global__ __launch_bounds__(256) void scan_k(
    const float* __restrict__ dt, const float* __restrict__ u,
    const float* __restrict__ xz, const float* __restrict__ dbc,
    const float* __restrict__ a_log, const float* __restrict__ dparam,
    float* __restrict__ y) {
  const int d = blockIdx.x * 256 + threadIdx.x;
  float a[S_], h[S_];
#pragma unroll
  for (int s = 0; s < S_; ++s) {
    a[s] = -__expf(a_log[d * S_ + s]);
    h[s] = 0.f;
  }
  const float Dv = dparam[d];
  __shared__ float sB[S_], sC[S_];
  for (int l = 0; l < L_; ++l) {
    __syncthreads();
    if (threadIdx.x < S_) {
      sB[threadIdx.x] = dbc[(size_t)l * DBCW + R_ + threadIdx.x];
      sC[threadIdx.x] = dbc[(size_t)l * DBCW + R_ + S_ + threadIdx.x];
    }
    __syncthreads();
    const float dtv = dt[(size_t)l * DI_ + d];
    const float uv = u[(size_t)l * DI_ + d];
    const float zv = xz[(size_t)l * (2 * DI_) + DI_ + d];
    const float dtu = dtv * uv;
    float yv = 0.f;
#pragma unroll
    for (int s = 0; s < S_; ++s) {
      const float dA = __expf(dtv * a[s]);
      h[s] = dA * h[s] + dtu * sB[s];
      yv += h[s] * sC[s];
    }
    yv += uv * Dv;
    yv *= zv / (1.f + __expf(-zv));
    y[(size_t)l * DI_ + d] = yv;
  }
}

// ---------------------------------------------------------------------------
// TDM: DMA a 2D W tile (nrows x KC f32, row pitch K) into LDS at lds_off,
// padding each 32-DWORD row with 4 DWORDs so LDS rows land on ROWP stride.
// D# per CDNA5 ISA §8.3/8.4: group0 = {count=1, lds_addr, global_addr, type=2},
// group1 = {data_size=4B, pad_enable, pad_interval=32dw, pad_amount=4dw,
//           tensor_dim0=KC, tensor_dim1=nrows, tile_dim0=KC, tile_dim1=nrows,
//           tensor_dim0_stride=K}. Groups 2/3 + extra group unused (2D tensor).
// ---------------------------------------------------------------------------
#define KC 32               // K-chunk (8 wmma k-steps)
#define ROWP 36             // padded LDS row stride (floats): 16B aligned rows,
                            // 36n mod 64 distinct for n=0..15 -> no bank conflicts

#if HAVE_TDM
__device__ __forceinline__ void tdm_stage_w(const float* gsrc,
                                            unsigned lds_off, int nrows,
                                            int K) {
  const unsigned long long ga = (unsigned long long)(uintptr_t)gsrc;
  v4u g0;
  g0[0] = 1u;                                   // count=1, user mode, no gather
  g0[1] = lds_off;                              // lds_addr (bytes)
  g0[2] = (unsigned)(ga & 0xFFFFFFFFu);         // global_addr[31:0]
  g0[3] = (unsigned)((ga >> 32) & 0x01FFFFFFu)  // global_addr[56:32]
          | (2u << 30);                         // type=2 ("image")
  v8i g1;
  g1[0] = (int)((2u << 16)     // data_size = 4 bytes
                | (1u << 20)   // pad_enable
                | (4u << 22)   // pad_interval code 4 -> every 32 DWORDs
                | (3u << 25)); // pad_amount  code 3 -> 4 DWORDs
  g1[1] = (int)(((unsigned)KC & 0xFFFFu) << 16);            // tensor_dim0 lo16
  g1[2] = (int)(((unsigned)nrows & 0xFFFFu) << 16);         // dim0 hi=0, dim1 lo
  g1[3] = (int)(((unsigned)KC & 0xFFFFu) << 16);            // dim1 hi=0, tile_dim0
  g1[4] = (int)((unsigned)nrows & 0xFFFFu);                 // tile_dim1 (tile_dim2=0)
  g1[5] = (int)(unsigned)K;                                 // tensor_dim0_stride lo32
  g1[6] = 0;                                                // stride hi, dim1_stride lo
  g1[7] = 0;
  const v4i z4 = {0, 0, 0, 0};
  const v8i z8 = {0, 0, 0, 0, 0, 0, 0, 0};
  __builtin_amdgcn_tensor_load_to_lds(g0, g1, z4, z4, z8, 0);
}

__device__ __forceinline__ void tdm_wait0() {
#if __has_builtin(__builtin_amdgcn_s_wait_tensorcnt)
  __builtin_amdgcn_s_wait_tensorcnt(0);
#else
  asm volatile("s_wait_tensorcnt 0x0" ::: "memory");
#endif
}
#endif  // HAVE_TDM

// ---------------------------------------------------------------------------
// WMMA f32 GEMM with double-buffered LDS staging of W (TDM when available).
//   out[M,N] = A[M,K] * W[N,K]^T  (+ epilogue)
//   EPI 0: none   EPI 1: += bias[col], softplus   EPI 2: += res[row*N+col]
//
// Block = 256 threads = 8 waves; block (bx,by) -> 128 x (16*NTILES) tile,
// wave w owns m-subtile (by*8+w)*16. All waves share one W N-strip: wave 0
// DMAs each NROWS x 32 W chunk into LDS via the Tensor Data Mover (TENSORcnt
// + s_wait_tensorcnt + barrier), overlapped with WMMA on the other buffer.
//
// Fragment layouts per CDNA5 ISA 7.12.2 (V_WMMA_F32_16X16X4_F32):
//   A 16x4 : lane m (0..15) holds {K=0,K=1}; lane m+16 {K=2,K=3}  -> v2f
//   B 4x16 : lane n (0..15) holds {K=0,K=1}; lane n+16 {K=2,K=3}  -> v2f
//   C 16x16: VGPR j -> row m0+j+8*half, col n0+(lane&15)          -> v8f
// ---------------------------------------------------------------------------
template <int NTILES, int EPI>
__global__ __launch_bounds__(256) void gemm_wmma_k(
    const float* __restrict__ A, const float* __restrict__ W,
    float* __restrict__ out, const float* __restrict__ extra,
    int M, int N, int K, int lda) {
  constexpr int NROWS = 16 * NTILES;
  __shared__ float sw[2][NROWS * ROWP];

  const int lane = threadIdx.x & 31;
  const int wave = threadIdx.x >> 5;
  const int half = lane >> 4;
  const int lrow = lane & 15;
  const int m0 = (blockIdx.y * 8 + wave) * 16;
  const int n0 = blockIdx.x * NROWS;

  const float* arow = A + (size_t)(m0 + lrow) * lda + 2 * half;
  const int nchunks = K / KC;

  v8f acc[NTILES];
#pragma unroll
  for (int t = 0; t < NTILES; ++t)
#pragma unroll
    for (int j = 0; j < 8; ++j) acc[t][j] = 0.f;

#if HAVE_TDM
  const float* wtile = W + (size_t)n0 * K;
  if (wave == 0) {
    tdm_stage_w(wtile, (unsigned)(uintptr_t)&sw[0][0], NROWS, K);
    tdm_wait0();
  }
  __syncthreads();
#else
  constexpr int VPT = (NROWS * KC) / (256 * 4);  // float4 per thread per stage
  auto stage = [&](int buf, int k0) {
#pragma unroll
    for (int v = 0; v < VPT; ++v) {
      const int vi = threadIdx.x + v * 256;  // 0..NROWS*8-1
      const int row = vi >> 3;
      const int c4 = (vi & 7) * 4;
      const float* gp = W + (size_t)(n0 + row) * K + k0 + c4;
      *(float4*)&sw[buf][row * ROWP + c4] = *(const float4*)gp;
      if (k0 + 2 * KC <= K - KC) __builtin_prefetch(gp + 2 * KC, 0, 1);
    }
  };
  stage(0, 0);
  __syncthreads();
#endif

  for (int c = 0; c < nchunks; ++c) {
    const int buf = c & 1;
#if HAVE_TDM
    if (c + 1 < nchunks && wave == 0)
      tdm_stage_w(wtile + (c + 1) * KC, (unsigned)(uintptr_t)&sw[buf ^ 1][0],
                  NROWS, K);
#else
    if (c + 1 < nchunks) stage(buf ^ 1, (c + 1) * KC);
#endif

    // batch the chunk's A fragments so their loads clause together
    v2f a[KC / 4];
#pragma unroll
    for (int kk = 0; kk < KC / 4; ++kk)
      a[kk] = *(const v2f*)(arow + c * KC + 4 * kk);

#pragma unroll
    for (int kk = 0; kk < KC / 4; ++kk) {
      v2f b[NTILES];  // gather all strips' fragments first -> one ds wait
#pragma unroll
      for (int t = 0; t < NTILES; ++t)
        b[t] = *(const v2f*)&sw[buf][(16 * t + lrow) * ROWP + 4 * kk + 2 * half];
#pragma unroll
      for (int t = 0; t < NTILES; ++t)
        acc[t] = __builtin_amdgcn_wmma_f32_16x16x4_f32(
            false, a[kk], false, b[t], (short)0, acc[t], false, false);
    }

#if HAVE_TDM
    if (c + 1 < nchunks && wave == 0) tdm_wait0();  // next buf resident in LDS
#endif
    __syncthreads();  // staging done; readers of buf done before reuse
  }

#pragma unroll
  for (int t = 0; t < NTILES; ++t) {
#pragma unroll
    for (int j = 0; j < 8; ++j) {
      const int row = m0 + j + 8 * half;
      const int col = n0 + 16 * t + lrow;
      float v = acc[t][j];
      if constexpr (EPI == 1) {  // bias + softplus (dt path)
        v += extra[col];
        v = (v > 20.f) ? v : log1pf(__expf(v));
      } else if constexpr (EPI == 2) {  // residual add (out_proj)
        v += extra[(size_t)row * N + col];
      }
      out[(size_t)row * N + col] = v;
    }
  }
}

// ---------------------------------------------------------------------------
extern "C" void kernel_launch(void* const* d_in, const int* in_sizes, int n_in,
                              void* d_out, int out_size, void* d_ws,
                              size_t ws_size, hipStream_t stream) {
  (void)in_sizes; (void)n_in; (void)out_size; (void)ws_size;
  const int* tokens = (const int*)d_in[0];
  const float* embed = (const float*)d_in[1];
  const float* norm_w = (const float*)d_in[2];
  const float* in_proj_w = (const float*)d_in[3];
  const float* conv_w = (const float*)d_in[4];
  const float* conv_b = (const float*)d_in[5];
  const float* x_proj_w = (const float*)d_in[6];
  const float* dt_proj_w = (const float*)d_in[7];
  const float* dt_proj_b = (const float*)d_in[8];
  const float* A_log = (const float*)d_in[9];
  const float* D_param = (const float*)d_in[10];
  const float* out_proj_w = (const float*)d_in[11];
  const float* norm_f_w = (const float*)d_in[12];
  const float* lm_head_w = (const float*)d_in[13];
  float* out = (float*)d_out;

  // Workspace layout (floats); total ~12.7M floats (~51 MB)
  float* ws = (float*)d_ws;
  float* X = ws;                              // L*D   residual stream
  float* XN = X + (size_t)L_ * D_;            // L*D   normed
  float* XZ = XN + (size_t)L_ * D_;           // L*2DI in_proj output
  float* U = XZ + (size_t)L_ * 2 * DI_;       // L*DI  conv+silu
  float* DBC = U + (size_t)L_ * DI_;          // L*96
  float* DT = DBC + (size_t)L_ * DBCW;        // L*DI
  float* Y = DT + (size_t)L_ * DI_;           // L*DI  scan output

  const dim3 blk(256);
  embed_gather_k<<<L_, blk, 0, stream>>>(tokens, embed, X);

  for (int i = 0; i < NL_; ++i) {
    rmsnorm_k<<<L_, blk, 0, stream>>>(X, norm_w + (size_t)i * D_, XN);

    // xz = xn @ in_proj_w^T   (M=1024, N=4096, K=1024)
    gemm_wmma_k<4, 0><<<dim3((2 * DI_) / 64, L_ / 128), blk, 0, stream>>>(
        XN, in_proj_w + (size_t)i * 2 * DI_ * D_, XZ, nullptr, L_, 2 * DI_,
        D_, D_);

    conv_silu_k<<<dim3(DI_ / 256, L_), blk, 0, stream>>>(
        XZ, conv_w + (size_t)i * DI_ * DC_, conv_b + (size_t)i * DI_, U);

    // dbc = u @ x_proj_w^T    (M=1024, N=96, K=2048)
    gemm_wmma_k<2, 0><<<dim3(DBCW / 32, L_ / 128), blk, 0, stream>>>(
        U, x_proj_w + (size_t)i * DBCW * DI_, DBC, nullptr, L_, DBCW, DI_,
        DI_);

    // dt = softplus(dbc[:, :R] @ dt_proj_w^T + b)  (M=1024,N=2048,K=64,lda=96)
    gemm_wmma_k<4, 1><<<dim3(DI_ / 64, L_ / 128), blk, 0, stream>>>(
        DBC, dt_proj_w + (size_t)i * DI_ * R_, DT,
        dt_proj_b + (size_t)i * DI_, L_, DI_, R_, DBCW);

    scan_k<<<DI_ / 256, blk, 0, stream>>>(
        DT, U, XZ, DBC, A_log + (size_t)i * DI_ * S_,
        D_param + (size_t)i * DI_, Y);

    // x = x + y @ out_proj_w^T  (M=1024, N=1024, K=2048), residual in-place
    gemm_wmma_k<4, 2><<<dim3(D_ / 64, L_ / 128), blk, 0, stream>>>(
        Y, out_proj_w + (size_t)i * D_ * DI_, X, X, L_, D_, DI_, DI_);
  }

  rmsnorm_k<<<L_, blk, 0, stream>>>(X, norm_f_w, XN);

  // logits = xn @ lm_head_w^T  (M=1024, N=32000, K=1024)
  gemm_wmma_k<4, 0><<<dim3(VOCAB / 64, L_ / 128), blk, 0, stream>>>(
      XN, lm_head_w, out, nullptr, L_, VOCAB, D_, D_);
}